// LocallyConnected2d_17059610100508
// MI455X (gfx1250) — compile-verified
//
#include <hip/hip_runtime.h>

typedef float v2f __attribute__((ext_vector_type(2)));
typedef float v8f __attribute__((ext_vector_type(8)));

// ---- problem constants ----
#define CIN  32
#define WW   64
#define COUT 32
#define OHH  62
#define OWW  62

// weight strides (elements): [1][o][c][h][w][k]
#define W_OS 1107072   // CIN*OH*OW*9
#define W_CS 34596     // OH*OW*9
#define W_HS 558       // OW*9
// x strides: [b][c][h][w]
#define X_BS 131072
#define X_CS 4096
#define X_HS 64
// out strides: [b][o][h][w]
#define O_BS 123008
#define O_OS 3844
#define O_HS 62

// ---- tiling ----
#define PX      8            // pixels (w) per block, 1 per wave32
#define CCH     4            // Cin per chunk (8 chunks total)
// B operand LDS layout: [px][cl][kpair(6)][o*2+parity] pair-interleaved so a
// WMMA B fragment (K pair) is a single 8B-aligned ds_load_b64.
#define BR2     68           // row stride: 64 + 4 pad (8B aligned, conflict-free)
#define CLSTR   (6*BR2)      // 408 floats per (px,cl)
#define PXSTR   (CCH*CLSTR)  // 1632 floats per px
#define BS_FLOATS (PX*PXSTR) // 13056 floats per B buffer
#define XSTR    17           // padded xT row stride
#define XT_FLOATS (CCH*3*10*XSTR)            // 2040 floats per xT buffer
#define XT2_OFF  (2*BS_FLOATS)               // xT buffers after both B buffers
#define SMEM_FLOATS (2*BS_FLOATS + 2*XT_FLOATS)  // 30192 floats = 120768 B

// asyncs issued per wave per chunk: 16 runs * 3 (weights) + 8 (x) = 56
#define ASYNC_PER_CHUNK 56

// ---- async global->LDS copy (CDNA5), with safe fallback ----
#if __has_builtin(__builtin_amdgcn_global_load_async_to_lds_b32)
typedef __attribute__((address_space(1))) int as1_int;
typedef __attribute__((address_space(3))) int as3_int;
static __device__ __forceinline__ void cp_g2l_b32(const float* g, float* l) {
  __builtin_amdgcn_global_load_async_to_lds_b32(
      (as1_int*)(void*)(float*)g, (as3_int*)(void*)l, 0, 0);
}
template <int N> static __device__ __forceinline__ void wait_async_le() {
#if __has_builtin(__builtin_amdgcn_s_wait_asynccnt)
  __builtin_amdgcn_s_wait_asynccnt(N);
#else
  asm volatile("s_wait_asynccnt %0" :: "i"(N) : "memory");
#endif
}
#else
static __device__ __forceinline__ void cp_g2l_b32(const float* g, float* l) {
  *l = __builtin_nontemporal_load(g);
}
template <int N> static __device__ __forceinline__ void wait_async_le() {}
#endif

__global__ __launch_bounds__(256)
void lc2d_wmma_f32(const float* __restrict__ x,
                   const float* __restrict__ wgt,
                   float* __restrict__ out) {
  __shared__ float smem[SMEM_FLOATS];

  const int tid = threadIdx.x;
  const int bid = blockIdx.x;          // 534 blocks (grid 62 * 69 / 8)
  const int h   = bid >> 3;
  const int w0  = (bid & 7) * PX;
  const int npx = (OWW - w0) < PX ? (OWW - w0) : PX;   // 8, or 6 in tail column

  const int lane = tid & 31;
  const int wid  = tid >> 5;           // wave id == pixel id
  const int hf   = lane >> 4;          // half-wave (K pair select)
  const int ln   = lane & 15;          // M (=b) for A, N (=o) for B/C/D

  // ---- zero-fill the K-pad slots of B (k = 9,10,11), BOTH buffers ----
  for (int i = 0; i < 24; ++i) {
    int f = tid + i * 256;                      // < 6144
    int buf = f >= 3072;
    int fo = f & 3071;
    int o = fo & 31, r = fo >> 5;               // r < 96
    int pc = r / 3, wh = r - pc * 3;
    int px = pc >> 2, cl = pc & 3;
    int kp  = (wh == 0) ? 4 : 5;
    int par = (wh == 1) ? 0 : 1;
    smem[buf * BS_FLOATS + px * PXSTR + cl * CLSTR + kp * BR2 + o * 2 + par] = 0.0f;
  }
  // ---- zero-fill xT out-of-range window columns (tail blocks), BOTH buffers
  if (w0 + 9 >= WW) {                           // only w0 == 56 (wq in {8,9})
    for (int i = 0; i < 3; ++i) {
      int f = tid + i * 256;                    // < 768
      int b  = f & 15;
      int r  = f >> 4;                          // < 48
      int wq = (WW - w0) + (r & 1);             // 8 or 9
      int r2 = r >> 1;                          // < 24
      int hp = r2 % 3;
      int r3 = r2 / 3;                          // < 8
      int cl = r3 & 3, buf = r3 >> 2;
      smem[XT2_OFF + buf * XT_FLOATS + ((cl * 3 + hp) * 10 + wq) * XSTR + b] = 0.0f;
    }
  }

  // ---- loop-invariant weight staging decomposition ----
  // wave owns 16 (o,cl) runs: cl = wid>>1 constant, o = (wid&1)*16 + r.
  // lane covers e = lane, lane+32, lane+64 within each 72-float run.
  // Invalid lanes redirect to a duplicate of lane-0/e=0's transfer (goff=0,
  // loff=0) so every wave issues a deterministic async count.
  const int clw  = wid >> 1;
  const int osub = (wid & 1) * 16;
  int ge[3], lo[3];
  #pragma unroll
  for (int v = 0; v < 3; ++v) {
    int e  = lane + v * 32;
    int px = e / 9;
    int k  = e - px * 9;
    bool valid = (px < npx) && (e < 72);
    ge[v] = valid ? e : 0;
    lo[v] = valid ? (px * PXSTR + (k >> 1) * BR2 + (k & 1)) : 0;
  }

  // ---- loop-invariant x staging decomposition (8 asyncs per thread) ----
  int xg[8], xl[8];
  #pragma unroll
  for (int i = 0; i < 8; ++i) {
    int f = tid + i * 256;
    bool v = (f < 1920);
    int ff = v ? f : 0;
    int wq = ff % 10; int r1 = ff / 10;
    int hp = r1 % 3;  int r2 = r1 / 3;
    int cl = r2 & 3;  int b  = r2 >> 2;
    v = v && (w0 + wq < WW);
    xg[i] = v ? (b * X_BS + cl * X_CS + hp * X_HS + wq) : 0;
    xl[i] = v ? (((cl * 3 + hp) * 10 + wq) * XSTR + b) : 0;
  }

  const float* wrow  = wgt + (size_t)h * W_HS + (size_t)w0 * 9;
  const float* xbase = x + (size_t)h * X_HS + w0;

  auto stage = [&](int c0, int buf) {
    // weights: 16 contiguous 72-float runs per wave, 3 asyncs per run
    const float* g = wrow + (size_t)(c0 + clw) * W_CS + (size_t)osub * W_OS;
    int lb = buf * BS_FLOATS + clw * CLSTR + osub * 2;
    #pragma unroll 4
    for (int r = 0; r < 16; ++r) {
      cp_g2l_b32(g + ge[0], &smem[lb + lo[0]]);
      cp_g2l_b32(g + ge[1], &smem[lb + lo[1]]);
      cp_g2l_b32(g + ge[2], &smem[lb + lo[2]]);
      g += W_OS;
      lb += 2;
    }
    // x window, transposed into xT[cl][h'][w''][b]
    const float* gx = xbase + (size_t)c0 * X_CS;
    int xb = XT2_OFF + buf * XT_FLOATS;
    #pragma unroll
    for (int i = 0; i < 8; ++i)
      cp_g2l_b32(gx + xg[i], &smem[xb + xl[i]]);
    // prefetch hint one chunk further ahead
    if (c0 + CCH < CIN) {
      int f = tid * 36;
      int run = f / 72;
      int e   = f - run * 72;
      int o   = run & 31, cl = run >> 5;
      __builtin_prefetch(wrow + (size_t)(c0 + CCH + cl) * W_CS +
                         (size_t)o * W_OS + e, 0, 1);
    }
  };

  v8f acc0 = {0.f,0.f,0.f,0.f,0.f,0.f,0.f,0.f};
  v8f acc1 = {0.f,0.f,0.f,0.f,0.f,0.f,0.f,0.f};

  stage(0, 0);   // prologue: chunk 0 into buffer 0

  for (int cc = 0; cc < CIN / CCH; ++cc) {
    const int buf = cc & 1;
    if (cc + 1 < CIN / CCH) {
      stage((cc + 1) * CCH, buf ^ 1);          // stream next chunk during compute
      wait_async_le<ASYNC_PER_CHUNK>();        // in-order: chunk cc fully landed
    } else {
      wait_async_le<0>();
    }
    __syncthreads();                           // chunk cc visible to all waves

    if (wid < npx) {                           // wave-uniform: EXEC all-ones
      const int xtb = XT2_OFF + buf * XT_FLOATS;
      #pragma unroll
      for (int cl = 0; cl < CCH; ++cl) {
        const int bbase = buf * BS_FLOATS + wid * PXSTR + cl * CLSTR;
        #pragma unroll
        for (int s = 0; s < 3; ++s) {
          int kk0 = s * 4 + hf * 2;            // lane's K pair: {0,1}/{2,3}
          int kk1 = kk0 + 1;
          int ka0 = kk0 > 8 ? 8 : kk0;         // clamp: matching B rows are zero
          int ka1 = kk1 > 8 ? 8 : kk1;
          int ki0 = ka0 / 3, kj0 = ka0 - ki0 * 3;
          int ki1 = ka1 / 3, kj1 = ka1 - ki1 * 3;
          v2f a;
          a.x = smem[xtb + ((cl * 3 + ki0) * 10 + wid + kj0) * XSTR + ln];
          a.y = smem[xtb + ((cl * 3 + ki1) * 10 + wid + kj1) * XSTR + ln];
          const int kp = s * 2 + hf;
          v2f b0 = *(const v2f*)&smem[bbase + kp * BR2 + ln * 2];
          v2f b1 = *(const v2f*)&smem[bbase + kp * BR2 + 32 + ln * 2];
          acc0 = __builtin_amdgcn_wmma_f32_16x16x4_f32(
                     false, a, false, b0, (short)0, acc0, false, false);
          acc1 = __builtin_amdgcn_wmma_f32_16x16x4_f32(
                     false, a, false, b1, (short)0, acc1, false, false);
        }
      }
    }
    __syncthreads();   // all waves done with buf before it is restaged
  }

  // ---- transpose D tiles through LDS for contiguous global stores ----
  if (wid < npx) {
    #pragma unroll
    for (int r = 0; r < 8; ++r) {
      int b = r + hf * 8;                      // C/D layout: vgpr r -> M=r (+8 hi)
      smem[(b * COUT + ln) * PX + wid]        = acc0[r];
      smem[(b * COUT + 16 + ln) * PX + wid]   = acc1[r];
    }
  }
  __syncthreads();
  for (int i = 0; i < 16; ++i) {
    int f  = tid + i * 256;                    // < 4096
    int px = f & 7;
    if (w0 + px < OWW) {
      int o = (f >> 3) & 31;
      int b = f >> 8;
      __builtin_nontemporal_store(
          smem[f],
          out + (size_t)b * O_BS + (size_t)o * O_OS + (size_t)h * O_HS + (w0 + px));
    }
  }
}

extern "C" void kernel_launch(void* const* d_in, const int* in_sizes, int n_in,
                              void* d_out, int out_size, void* d_ws, size_t ws_size,
                              hipStream_t stream) {
  (void)in_sizes; (void)n_in; (void)out_size; (void)d_ws; (void)ws_size;
  const float* x   = (const float*)d_in[0];
  const float* wgt = (const float*)d_in[1];
  float*       out = (float*)d_out;
  dim3 grid(OHH * (OWW + PX - 1) / PX);   // 62 * 69 / 8 = 534 blocks
  dim3 block(256);                        // 8 wave32 waves
  lc2d_wmma_f32<<<grid, block, 0, stream>>>(x, wgt, out);
}